// GPNet_23785528885974
// MI455X (gfx1250) — compile-verified
//
#include <hip/hip_runtime.h>

// GPNet unrolled beamforming for MI455X (gfx1250).
// f32 WMMA (v_wmma_f32_16x16x4_f32) for all matmuls, one WG (4 wave32) per batch,
// register-blocked tile rows, TDM (tensor_load_to_lds) staging of U[t] with
// hardware LDS padding, ~92KB LDS -> 3 workgroups per WGP.

#define NT 64          // Nt antennas
#define KU 40          // K users
#define KP 48          // K padded to WMMA tile multiple
#define LDK 49         // odd LDS stride for 48-col arrays (conflict-free transposed reads)
#define LDU 65         // odd LDS stride for 64-col U (TDM pad: 1 DWORD every 64)
#define NLAYERS 7
#define NOISE_C 0.02f
#define ALPHA 1.4426950408889634f  // 1/ln(2)
#define NTHREADS 128

typedef float v2f __attribute__((ext_vector_type(2)));
typedef float v8f __attribute__((ext_vector_type(8)));
typedef unsigned int u32x4 __attribute__((ext_vector_type(4)));
typedef int i32x4 __attribute__((ext_vector_type(4)));
typedef int i32x8 __attribute__((ext_vector_type(8)));

#if __has_builtin(__builtin_amdgcn_wmma_f32_16x16x4_f32)
#define HAVE_WMMA_F32 1
#endif
#if __has_builtin(__builtin_amdgcn_tensor_load_to_lds) && \
    __has_builtin(__builtin_amdgcn_s_wait_tensorcnt)
#define HAVE_TDM 1
#endif

__device__ __forceinline__ v8f wmma_f32(v2f a, v2f b, v8f c) {
#ifdef HAVE_WMMA_F32
  return __builtin_amdgcn_wmma_f32_16x16x4_f32(false, a, false, b, (short)0, c,
                                               false, false);
#else
  for (int i = 0; i < 8; ++i) c[i] += a.x * b.x + a.y * b.y;
  return c;
#endif
}

// A-operand 16x4, column-major source (X^T): A[m][c] = X[c0+c][rowbase+m]
__device__ __forceinline__ v2f fragA_cm(const float* X, int ld, int rowbase,
                                        int c0, int lane) {
  int m = lane & 15, cb = (lane >> 4) << 1;
  v2f a;
  a.x = X[(c0 + cb + 0) * ld + rowbase + m];
  a.y = X[(c0 + cb + 1) * ld + rowbase + m];
  return a;
}
// A-operand 16x4 row-major: A[m][c] = X[rowbase+m][c0+c]
__device__ __forceinline__ v2f fragA_rm(const float* X, int ld, int rowbase,
                                        int c0, int lane) {
  int m = lane & 15, cb = (lane >> 4) << 1;
  v2f a;
  a.x = X[(rowbase + m) * ld + c0 + cb + 0];
  a.y = X[(rowbase + m) * ld + c0 + cb + 1];
  return a;
}
// B-operand 4x16 row-major: B[c][n] = X[c0+c][colbase+n]
__device__ __forceinline__ v2f fragB_rm(const float* X, int ld, int colbase,
                                        int c0, int lane) {
  int n = lane & 15, cb = (lane >> 4) << 1;
  v2f b;
  b.x = X[(c0 + cb + 0) * ld + colbase + n];
  b.y = X[(c0 + cb + 1) * ld + colbase + n];
  return b;
}
// C/D 16x16: VGPR r -> row rowbase + r + 8*(lane>=16), col colbase + (lane&15)
__device__ __forceinline__ void storeCD(float* X, int ld, int rowbase,
                                        int colbase, v8f acc, int lane) {
  int n = lane & 15, ro = (lane >> 4) << 3;
#pragma unroll
  for (int r = 0; r < 8; ++r)
    X[(rowbase + ro + r) * ld + colbase + n] = acc[r];
}

#define V8F_ZERO {0.f,0.f,0.f,0.f,0.f,0.f,0.f,0.f}

extern "C" __global__ void __launch_bounds__(NTHREADS)
gpnet_kernel(const float* __restrict__ Hr, const float* __restrict__ Hi,
             const float* __restrict__ eta, const float* __restrict__ U,
             const float* __restrict__ bvec, float* __restrict__ out) {
  extern __shared__ float smem[];
  float* sHr = smem;                       // 64*49
  float* sHi = sHr + NT * LDK;
  float* sWr = sHi + NT * LDK;
  float* sWi = sWr + NT * LDK;
  float* sGr = sWi + NT * LDK;
  float* sGi = sGr + NT * LDK;
  float* sAr = sGi + NT * LDK;             // 48*49 (A, then M in place)
  float* sAi = sAr + KP * LDK;
  float* sU  = sAr;                        // ALIAS: U[t] (64*65) lives in sA region,
                                           // only between Phase G and Phase Ug
  float* red = sAi + KP * LDK;             // 4 (per-wave partials)
  float* sPb = red + 4;                    // 1

  const int tid  = threadIdx.x;
  const int lane = tid & 31;
  const int wave = tid >> 5;
  const int bidx = blockIdx.x;
  const float* gHr = Hr + (size_t)bidx * NT * KU;
  const float* gHi = Hi + (size_t)bidx * NT * KU;
  __builtin_prefetch(gHr, 0, 1);
  __builtin_prefetch(gHi, 0, 1);

  // Stage H (zero-padded to 48 cols), zero W.
  for (int idx = tid; idx < NT * KP; idx += NTHREADS) {
    int n = idx / KP, k = idx - n * KP;
    float hr = 0.f, hi = 0.f;
    if (k < KU) { hr = gHr[n * KU + k]; hi = gHi[n * KU + k]; }
    sHr[n * LDK + k] = hr;  sHi[n * LDK + k] = hi;
    sWr[n * LDK + k] = 0.f; sWi[n * LDK + k] = 0.f;
  }
  __syncthreads();

  for (int t = 0; t < NLAYERS; ++t) {
    const float eta_t = eta[t];
    const float* bt = bvec + t * NT * KU;
    const float* Ut = U + t * NT * NT;

    // ---- Phase A: A = H^H W (48x48 complex). Waves 0-2: one 16-row tile each,
    // ---- 3 column tiles register-blocked (A-frags reused 3x). Wave 3: ||W||^2.
    if (wave < 3) {
      const int ti = wave * 16;
      v8f aR[3] = {V8F_ZERO, V8F_ZERO, V8F_ZERO};
      v8f aI[3] = {V8F_ZERO, V8F_ZERO, V8F_ZERO};
      for (int c0 = 0; c0 < NT; c0 += 4) {
        v2f ahr = fragA_cm(sHr, LDK, ti, c0, lane);
        v2f ahi = fragA_cm(sHi, LDK, ti, c0, lane);
        v2f nhi = -ahi;                    // f32 WMMA has no A-neg modifier
#pragma unroll
        for (int tj = 0; tj < 3; ++tj) {
          v2f bwr = fragB_rm(sWr, LDK, tj * 16, c0, lane);
          v2f bwi = fragB_rm(sWi, LDK, tj * 16, c0, lane);
          aR[tj] = wmma_f32(ahr, bwr, aR[tj]);   // Hr^T Wr
          aR[tj] = wmma_f32(ahi, bwi, aR[tj]);   // + Hi^T Wi
          aI[tj] = wmma_f32(ahr, bwi, aI[tj]);   // Hr^T Wi
          aI[tj] = wmma_f32(nhi, bwr, aI[tj]);   // - Hi^T Wr
        }
      }
#pragma unroll
      for (int tj = 0; tj < 3; ++tj) {
        storeCD(sAr, LDK, ti, tj * 16, aR[tj], lane);
        storeCD(sAi, LDK, ti, tj * 16, aI[tj], lane);
      }
    } else {
      // wave 3: power reduction ||W||_F^2 (W is stable during Phase A)
      float p = 0.f;
      for (int idx = lane; idx < NT * KP; idx += 32) {
        int n = idx / KP, k = idx - n * KP;
        float wr = sWr[n * LDK + k], wi = sWi[n * LDK + k];
        p += wr * wr + wi * wi;
      }
      for (int off = 16; off > 0; off >>= 1) p += __shfl_xor(p, off, 32);
      if (lane == 0) *sPb = p;
    }
    __syncthreads();

    // ---- Phase C+M fused: thread k computes its row's SINR coefficients and
    // ---- scales row k of A into M in place. Rows/cols >=KU are already zero.
    if (tid < KU) {
      const int k = tid;
      float T = NOISE_C;
      for (int l = 0; l < KU; ++l) {
        float ar = sAr[k * LDK + l], ai = sAi[k * LDK + l];
        T += ar * ar + ai * ai;
      }
      float dr = sAr[k * LDK + k], di = sAi[k * LDK + k];
      float Un = T - (dr * dr + di * di);        // I_k + noise
      float cd = ALPHA / T;
      float co = ALPHA * (1.f / T - 1.f / Un);
      for (int l = 0; l < KP; ++l) {
        float c = (l == k) ? cd : co;
        sAr[k * LDK + l] *= c;
        sAi[k * LDK + l] *= c;
      }
    }
    __syncthreads();

    const float gamma = 0.04f * (*sPb - 1.0f);

    // ---- Phase G: G = 2*H*M - gamma*W. Wave w owns row tile w (4x16 rows),
    // ---- 3 column tiles register-blocked. Inner dim 48.
    {
      const int ti = wave * 16;
      v8f aR[3] = {V8F_ZERO, V8F_ZERO, V8F_ZERO};
      v8f aI[3] = {V8F_ZERO, V8F_ZERO, V8F_ZERO};
      for (int c0 = 0; c0 < KP; c0 += 4) {
        v2f ahr = fragA_rm(sHr, LDK, ti, c0, lane);
        v2f ahi = fragA_rm(sHi, LDK, ti, c0, lane);
        v2f nhi = -ahi;
#pragma unroll
        for (int tj = 0; tj < 3; ++tj) {
          v2f bmr = fragB_rm(sAr, LDK, tj * 16, c0, lane);
          v2f bmi = fragB_rm(sAi, LDK, tj * 16, c0, lane);
          aR[tj] = wmma_f32(ahr, bmr, aR[tj]);   // Hr Mr
          aR[tj] = wmma_f32(nhi, bmi, aR[tj]);   // - Hi Mi
          aI[tj] = wmma_f32(ahr, bmi, aI[tj]);   // Hr Mi
          aI[tj] = wmma_f32(ahi, bmr, aI[tj]);   // + Hi Mr
        }
      }
      int n0 = lane & 15, ro = (lane >> 4) << 3;
#pragma unroll
      for (int tj = 0; tj < 3; ++tj)
#pragma unroll
        for (int r = 0; r < 8; ++r) {
          int a = (ti + ro + r) * LDK + (tj * 16 + n0);
          sGr[a] = 2.f * aR[tj][r] - gamma * sWr[a];
          sGi[a] = 2.f * aI[tj][r] - gamma * sWi[a];
        }
    }
    __syncthreads();   // sA/M dead from here -> sU may overwrite it

    // ---- Stage U[t] into sU (stride 65) via Tensor Data Mover ----
#ifdef HAVE_TDM
    {
      // D# group0: count=1, lds_addr, 57-bit global addr, type=2
      u32x4 g0;
      g0[0] = 1u;                                  // count = 1 valid descriptor
      g0[1] = (unsigned)(size_t)sU;                // LDS byte offset
      unsigned long long ga = (unsigned long long)(size_t)Ut;
      g0[2] = (unsigned)ga;
      g0[3] = (unsigned)((ga >> 32) & 0x01FFFFFFu) | (2u << 30);  // type=2
      // D# group1: data_size=4B, pad_enable, pad every 64 DWORDs by 1 DWORD
      // (reproduces the stride-65 conflict-free layout), 64x64 tile == tensor.
      i32x8 g1;
      g1[0] = (int)((2u << 16) | (1u << 20) | (5u << 22)); // dsize|pad_en|pad_int=64
      g1[1] = (int)(64u << 16);                    // tensor_dim0 = 64 (lo16)
      g1[2] = (int)(64u << 16);                    // dim0 hi=0 | tensor_dim1 = 64
      g1[3] = (int)(64u << 16);                    // dim1 hi=0 | tile_dim0 = 64
      g1[4] = (int)(64u);                          // tile_dim1 = 64 | tile_dim2 = 0
      g1[5] = (int)(64u);                          // tensor_dim0_stride = 64
      g1[6] = 0; g1[7] = 0;
      i32x4 gz4 = {0, 0, 0, 0};
      i32x8 gz8 = {0, 0, 0, 0, 0, 0, 0, 0};
      if (wave == 0) {                             // uniform branch: one TDM op/WG
        // 6-arg toolchain variant: (g0, g1, g2, g3, g4, cpol)
        __builtin_amdgcn_tensor_load_to_lds(g0, g1, gz4, gz4, gz8, 0);
        __builtin_amdgcn_s_wait_tensorcnt((short)0);
      }
    }
#else
    for (int idx = tid; idx < NT * NT; idx += NTHREADS)
      sU[(idx >> 6) * LDU + (idx & 63)] = Ut[idx];
#endif
    __syncthreads();

    // ---- Phase Ug: Ug = U_t G; W <- W - eta*(1+i)*Ug + b_t. Wave w: row tile w.
    {
      const int ti = wave * 16;
      v8f aR[3] = {V8F_ZERO, V8F_ZERO, V8F_ZERO};
      v8f aI[3] = {V8F_ZERO, V8F_ZERO, V8F_ZERO};
      for (int c0 = 0; c0 < NT; c0 += 4) {
        v2f au = fragA_rm(sU, LDU, ti, c0, lane);
#pragma unroll
        for (int tj = 0; tj < 3; ++tj) {
          v2f bgr = fragB_rm(sGr, LDK, tj * 16, c0, lane);
          v2f bgi = fragB_rm(sGi, LDK, tj * 16, c0, lane);
          aR[tj] = wmma_f32(au, bgr, aR[tj]);
          aI[tj] = wmma_f32(au, bgi, aI[tj]);
        }
      }
      int n0 = lane & 15, ro = (lane >> 4) << 3;
#pragma unroll
      for (int tj = 0; tj < 3; ++tj)
#pragma unroll
        for (int r = 0; r < 8; ++r) {
          int row = ti + ro + r, col = tj * 16 + n0;
          int a = row * LDK + col;
          float ur = aR[tj][r], ui = aI[tj][r];
          float wr = sWr[a] - eta_t * (ur - ui);
          float wi = sWi[a] - eta_t * (ui + ur);
          if (col < KU) wr += bt[row * KU + col];
          sWr[a] = wr; sWi[a] = wi;
        }
    }
    __syncthreads();
  }

  // ---- Final power normalization + interleaved complex64 writeout ----
  {
    float p = 0.f;
    for (int idx = tid; idx < NT * KP; idx += NTHREADS) {
      int n = idx / KP, k = idx - n * KP;
      float wr = sWr[n * LDK + k], wi = sWi[n * LDK + k];
      p += wr * wr + wi * wi;
    }
    for (int off = 16; off > 0; off >>= 1) p += __shfl_xor(p, off, 32);
    if (lane == 0) red[wave] = p;
  }
  __syncthreads();
  if (tid == 0) {
    float s = red[0] + red[1] + red[2] + red[3];
    *sPb = 1.0f / (sqrtf(s) + 1e-6f);   // sqrt(P)=1
  }
  __syncthreads();
  const float scale = *sPb;
  float* gout = out + (size_t)bidx * NT * KU * 2;
  for (int idx = tid; idx < NT * KU; idx += NTHREADS) {
    int n = idx / KU, k = idx - n * KU;
    gout[2 * idx + 0] = sWr[n * LDK + k] * scale;
    gout[2 * idx + 1] = sWi[n * LDK + k] * scale;
  }
}

extern "C" void kernel_launch(void* const* d_in, const int* in_sizes, int n_in,
                              void* d_out, int out_size, void* d_ws,
                              size_t ws_size, hipStream_t stream) {
  const float* Hr  = (const float*)d_in[0];
  const float* Hi  = (const float*)d_in[1];
  const float* eta = (const float*)d_in[2];
  const float* U   = (const float*)d_in[3];
  const float* bv  = (const float*)d_in[4];
  float* out = (float*)d_out;

  const int B = in_sizes[0] / (NT * KU);  // 2048
  // 6*(64*49) + 2*(48*49) + 4 + 1 floats  (sU aliases the sA region)
  const size_t smem_floats = 6 * NT * LDK + 2 * KP * LDK + 4 + 1;
  const size_t smem_bytes = smem_floats * sizeof(float);

  gpnet_kernel<<<dim3(B), dim3(NTHREADS), smem_bytes, stream>>>(Hr, Hi, eta, U,
                                                                bv, out);
}